// MultiBoxDetectionLoss_29798483100028
// MI455X (gfx1250) — compile-verified
//
#include <hip/hip_runtime.h>
#include <hip/hip_bf16.h>

// Problem constants from the reference
#define NB 64
#define NA 8732
#define NC 81
#define NS 16384   // next pow2 >= NA, for bitonic sort (64 KB LDS, fits 320KB/WGP)

typedef float v2f __attribute__((ext_vector_type(2)));
typedef float v8f __attribute__((ext_vector_type(8)));

// ---------------------------------------------------------------------------
// Kernel 1: per-anchor cross-entropy  c_loss[b,a] = logsumexp(scores) - score[label]
// One wave (32 lanes) per anchor row: lanes span classes -> fully coalesced
// 128B loads. This is the bandwidth-bound part (181 MB stream).
// ---------------------------------------------------------------------------
__global__ __launch_bounds__(256)
void closs_kernel(const float* __restrict__ scores,
                  const int*   __restrict__ labels,
                  float*       __restrict__ closs,
                  int nba) {
    const int lane  = threadIdx.x & 31;
    const int wave  = blockIdx.x * (blockDim.x >> 5) + (threadIdx.x >> 5);
    const int nwave = gridDim.x * (blockDim.x >> 5);

    for (int idx = wave; idx < nba; idx += nwave) {
        const float* row = scores + (size_t)idx * NC;
        if (idx + nwave < nba) {
            // CDNA5 global_prefetch_b8 hint for next row this wave will touch
            __builtin_prefetch(scores + (size_t)(idx + nwave) * NC, 0, 0);
        }
        float x0 = row[lane];
        float x1 = row[lane + 32];
        float x2 = (lane < NC - 64) ? row[lane + 64] : -3.0e38f;

        float m = fmaxf(fmaxf(x0, x1), x2);
        #pragma unroll
        for (int off = 16; off > 0; off >>= 1)
            m = fmaxf(m, __shfl_xor(m, off, 32));

        float e = __expf(x0 - m) + __expf(x1 - m);
        if (lane < NC - 64) e += __expf(x2 - m);
        #pragma unroll
        for (int off = 16; off > 0; off >>= 1)
            e += __shfl_xor(e, off, 32);

        if (lane == 0) {
            int lbl = labels[idx];
            closs[idx] = m + __logf(e) - row[lbl];
        }
    }
}

// ---------------------------------------------------------------------------
// Block reduction helper (wave32 shfl + tiny LDS scratch)
// ---------------------------------------------------------------------------
__device__ __forceinline__ float blockReduceSum(float v, float* red) {
    #pragma unroll
    for (int off = 16; off > 0; off >>= 1) v += __shfl_xor(v, off, 32);
    const int wid  = threadIdx.x >> 5;
    const int lane = threadIdx.x & 31;
    __syncthreads();
    if (lane == 0) red[wid] = v;
    __syncthreads();
    const int nw = blockDim.x >> 5;
    if (wid == 0) {
        float w = (lane < nw) ? red[lane] : 0.0f;
        #pragma unroll
        for (int off = 16; off > 0; off >>= 1) w += __shfl_xor(w, off, 32);
        if (lane == 0) red[0] = w;
    }
    __syncthreads();
    return red[0];
}

__device__ __forceinline__ float smoothL1(float d) {
    float ad = fabsf(d);
    return (ad < 1.0f) ? 0.5f * d * d : ad - 0.5f;
}

// ---------------------------------------------------------------------------
// Kernel 2: one block per image.
//   - pos_count, sum(c_loss*mask), sum(smoothL1*mask)
//   - conf_neg into LDS, pad to 16384, bitonic sort descending,
//     sum top K = min(3*pos_count, A)  (exact hard-negative mining)
// ---------------------------------------------------------------------------
__global__ __launch_bounds__(1024)
void per_image_kernel(const float*         __restrict__ closs,
                      const unsigned char* __restrict__ mask,
                      const float*         __restrict__ boxes,
                      const float*         __restrict__ gt_boxes,
                      float*               __restrict__ img_out) {
    extern __shared__ float s[];          // NS floats = 64 KB
    __shared__ float red[32];

    const int b   = blockIdx.x;
    const int tid = threadIdx.x;
    const int T   = blockDim.x;

    const float*         cl = closs + (size_t)b * NA;
    const unsigned char* mk = mask  + (size_t)b * NA;
    const float4*        bx = (const float4*)boxes    + (size_t)b * NA;
    const float4*        gb = (const float4*)gt_boxes + (size_t)b * NA;

    float pos_cnt = 0.0f, pos_sum = 0.0f, loc_sum = 0.0f;

    for (int a = tid; a < NA; a += T) {
        float c = cl[a];
        bool  m = (mk[a] != 0);
        s[a] = m ? 0.0f : c;                 // conf_neg
        if (m) {
            pos_cnt += 1.0f;
            pos_sum += c;
            float4 p = bx[a];
            float4 g = gb[a];
            loc_sum += smoothL1(p.x - g.x) + smoothL1(p.y - g.y) +
                       smoothL1(p.z - g.z) + smoothL1(p.w - g.w);
        }
    }
    for (int a = NA + tid; a < NS; a += T) s[a] = -1.0f;   // pad below all (c_loss >= 0)
    __syncthreads();

    pos_cnt = blockReduceSum(pos_cnt, red);
    pos_sum = blockReduceSum(pos_sum, red);
    loc_sum = blockReduceSum(loc_sum, red);

    const int K = min(3 * (int)(pos_cnt + 0.5f), NA);

    // Bitonic sort, descending
    for (int k = 2; k <= NS; k <<= 1) {
        for (int j = k >> 1; j > 0; j >>= 1) {
            __syncthreads();
            for (int i = tid; i < NS; i += T) {
                int p = i ^ j;
                if (p > i) {
                    float a0 = s[i], a1 = s[p];
                    bool up = ((i & k) == 0);
                    if (up ? (a0 < a1) : (a0 > a1)) { s[i] = a1; s[p] = a0; }
                }
            }
        }
    }
    __syncthreads();

    float nsum = 0.0f;
    for (int i = tid; i < K; i += T) nsum += s[i];
    nsum = blockReduceSum(nsum, red);

    if (tid == 0) {
        img_out[0 * NB + b] = loc_sum;
        img_out[1 * NB + b] = pos_sum;
        img_out[2 * NB + b] = nsum;
        img_out[3 * NB + b] = pos_cnt;
    }
}

// ---------------------------------------------------------------------------
// Kernel 3: single wave. Cross-image reduction of the four 64-element arrays
// via V_WMMA_F32_16X16X4_F32: A = 16x4 tile holding the 64 values (full f32
// precision), B = ones(4x16)  =>  D[m][n] = row-sum of A. One wmma per array.
// ---------------------------------------------------------------------------
__device__ __forceinline__ float wmma_sum64(const float* __restrict__ data, int lane) {
    const int m  = lane & 15;
    const int hi = lane >> 4;                 // 0: K=0,1   1: K=2,3
    v2f a;
    a.x = data[4 * m + 2 * hi + 0];
    a.y = data[4 * m + 2 * hi + 1];
    v2f ones;
    ones.x = 1.0f; ones.y = 1.0f;
    v8f c = {};
    c = __builtin_amdgcn_wmma_f32_16x16x4_f32(false, a, false, ones,
                                              (short)0, c, false, false);
    // VGPR r: lanes 0-15 -> row r, lanes 16-31 -> row r+8 (replicated over n)
    float r = c[0] + c[1] + c[2] + c[3] + c[4] + c[5] + c[6] + c[7];
    r += __shfl_xor(r, 16, 32);               // rows 0-7 + rows 8-15
    return r;
}

__global__ __launch_bounds__(32)
void finalize_kernel(const float* __restrict__ img, float* __restrict__ out) {
    const int lane = threadIdx.x;
    float loc = wmma_sum64(img + 0 * NB, lane);
    float pos = wmma_sum64(img + 1 * NB, lane);
    float neg = wmma_sum64(img + 2 * NB, lane);
    float cnt = wmma_sum64(img + 3 * NB, lane);
    if (lane == 0) {
        float N         = fmaxf(1.0f, cnt);
        float loc_loss  = loc / N;
        float conf_loss = (pos + neg) / N;
        out[0] = loc_loss + conf_loss;
        out[1] = loc_loss;
        out[2] = conf_loss;
    }
}

// ---------------------------------------------------------------------------
extern "C" void kernel_launch(void* const* d_in, const int* in_sizes, int n_in,
                              void* d_out, int out_size, void* d_ws, size_t ws_size,
                              hipStream_t stream) {
    const float*         scores    = (const float*)d_in[0];
    const float*         boxes     = (const float*)d_in[1];
    const int*           gt_labels = (const int*)d_in[2];
    const float*         gt_boxes  = (const float*)d_in[3];
    const unsigned char* mask      = (const unsigned char*)d_in[4]; // jax bool = 1 byte

    float* closs = (float*)d_ws;            // NB*NA floats
    float* img   = closs + (size_t)NB * NA; // 4 * NB floats

    closs_kernel<<<2048, 256, 0, stream>>>(scores, gt_labels, closs, NB * NA);
    per_image_kernel<<<NB, 1024, NS * sizeof(float), stream>>>(closs, mask, boxes,
                                                               gt_boxes, img);
    finalize_kernel<<<1, 32, 0, stream>>>(img, (float*)d_out);
}